// KNNGraph_5549097746962
// MI455X (gfx1250) — compile-verified
//
#include <hip/hip_runtime.h>
#include <hip/hip_bf16.h>

// CDNA5 (gfx1250) brute-force kNN via V_WMMA_F32_16X16X4_F32.
// Score s[m][n] = |c_n|^2 - 2*q_m.c_n  (== |q_m - c_n|^2 - |q_m|^2, a per-row
// constant offset, so per-row ordering of scores == ordering of distances).
// A-fragment = -2 * query coords (K=0..2, K=3 padded 0), B = candidate coords,
// C seeded with broadcast |c|^2  ->  D = C + A*B gives the score directly.

typedef __attribute__((ext_vector_type(2))) float v2f;
typedef __attribute__((ext_vector_type(8))) float v8f;

constexpr int K_NN        = 16;    // neighbors per query
constexpr int CHUNK       = 1024;  // candidates staged in LDS per block iteration
constexpr int WAVES       = 8;     // 256 threads / wave32
constexpr int SCORE_STRIDE = 36;   // row stride in floats: conflict-free + 16B aligned

// Sorted-ascending insert into register-resident top-K list (fully unrolled).
#define INSERT(dval, ival)                                                    \
  do {                                                                        \
    float _d = (dval);                                                        \
    int _i = (ival);                                                          \
    if (_d < dist[K_NN - 1]) {                                                \
      dist[K_NN - 1] = _d;                                                    \
      idx[K_NN - 1] = _i;                                                     \
      _Pragma("unroll") for (int _j = K_NN - 1; _j > 0; --_j) {               \
        if (dist[_j] < dist[_j - 1]) {                                        \
          float _td = dist[_j]; dist[_j] = dist[_j - 1]; dist[_j - 1] = _td;  \
          int _ti = idx[_j];   idx[_j] = idx[_j - 1];   idx[_j - 1] = _ti;    \
        }                                                                     \
      }                                                                       \
    }                                                                         \
  } while (0)

__global__ __launch_bounds__(256) void knn_wmma_kernel(
    const float* __restrict__ pos, int n,
    int* __restrict__ out_nbr, int* __restrict__ out_ctr)
{
  __shared__ __align__(16) float4 cand[CHUNK];                    // {x,y,z,|c|^2}
  __shared__ __align__(16) float  score[WAVES][16][SCORE_STRIDE]; // 16x16 tiles
  __shared__ __align__(16) float  fin_d[WAVES][16][K_NN];
  __shared__ __align__(16) int    fin_i[WAVES][16][K_NN];

  const int tid  = threadIdx.x;
  const int wave = tid >> 5;
  const int lane = tid & 31;
  const int half = lane >> 4;   // 0: lanes 0-15, 1: lanes 16-31
  const int sub  = lane & 15;
  const int qbase = blockIdx.x * 16;   // 16 queries per block, shared by all waves

  // ---- A fragment: 16x4 f32, layout lanes 0-15 = {K0,K1}, lanes 16-31 = {K2,K3}
  const float* pq = pos + 3 * (qbase + sub);
  const float qx = pq[0], qy = pq[1], qz = pq[2];
  v2f afrag;
  afrag.x = -2.0f * (half ? qz : qx);
  afrag.y = half ? 0.0f : (-2.0f * qy);   // K3 pad = 0

  // Register-resident sorted top-K (ascending score) per owner lane.
  float dist[K_NN];
  int   idx[K_NN];
#pragma unroll
  for (int j = 0; j < K_NN; ++j) { dist[j] = 3.4e38f; idx[j] = 0; }

  for (int cb = 0; cb < n; cb += CHUNK) {
    __syncthreads();
    // Cooperative stage: candidate coords + squared norm into LDS.
    for (int i = tid; i < CHUNK; i += 256) {
      const int p = cb + i;
      if (p < n) {
        const float* pp = pos + 3 * p;
        const float x = pp[0], y = pp[1], z = pp[2];
        cand[i] = make_float4(x, y, z, x * x + y * y + z * z);
      }
    }
    __syncthreads();

    // Waves split the candidate tiles of this chunk.
    for (int t = wave; t < CHUNK / 16; t += WAVES) {
      const float4 cp = cand[t * 16 + sub];          // broadcast across halves
      v2f bfrag;                                     // B: 4x16, same striping as A
      bfrag.x = half ? cp.z : cp.x;
      bfrag.y = half ? 0.0f : cp.y;
      const float sq = cp.w;
      v8f acc = {sq, sq, sq, sq, sq, sq, sq, sq};    // C[m][n] = |c_n|^2

      // D = |c|^2 - 2 q.c   (one instruction = 256 pair scores)
      acc = __builtin_amdgcn_wmma_f32_16x16x4_f32(
          false, afrag, false, bfrag, (short)0, acc, false, false);

      // Scatter C layout (vgpr r, lane -> row r+8*half, col sub) to row-major LDS.
#pragma unroll
      for (int r = 0; r < 8; ++r)
        score[wave][r + 8 * half][sub] = acc[r];

      // Owner lanes (0-15) each maintain top-K for one query row.
      if (half == 0) {
        const int cidx0 = cb + t * 16;
        const float* row = &score[wave][sub][0];
#pragma unroll
        for (int q4 = 0; q4 < 4; ++q4) {
          const float4 s4 = *(const float4*)(row + q4 * 4);
          const float mn = fminf(fminf(s4.x, s4.y), fminf(s4.z, s4.w));
          if (mn < dist[K_NN - 1]) {                 // rare path after warm-up
            const float sv[4] = {s4.x, s4.y, s4.z, s4.w};
#pragma unroll
            for (int e = 0; e < 4; ++e)
              INSERT(sv[e], cidx0 + q4 * 4 + e);
          }
        }
      }
    }
  }

  // ---- Cross-wave merge: dump per-wave partial lists, wave 0 merges.
  if (half == 0) {
#pragma unroll
    for (int j = 0; j < K_NN; ++j) {
      fin_d[wave][sub][j] = dist[j];
      fin_i[wave][sub][j] = idx[j];
    }
  }
  __syncthreads();

  if (wave == 0 && half == 0) {
    for (int w = 1; w < WAVES; ++w) {
#pragma unroll
      for (int j = 0; j < K_NN; ++j) {
        const float d = fin_d[w][sub][j];
        if (d >= dist[K_NN - 1]) break;              // sublists are sorted
        INSERT(d, fin_i[w][sub][j]);
      }
    }
    const int q = qbase + sub;
#pragma unroll
    for (int j = 0; j < K_NN; ++j) {
      out_nbr[q * K_NN + j] = idx[j];                // ascending distance
      out_ctr[q * K_NN + j] = q;
    }
  }
}

extern "C" void kernel_launch(void* const* d_in, const int* in_sizes, int n_in,
                              void* d_out, int out_size, void* d_ws, size_t ws_size,
                              hipStream_t stream) {
  const float* pos = (const float*)d_in[0];
  const int n = in_sizes[0] / 3;                     // 16384 points
  int* out_nbr = (int*)d_out;                        // int32 indices, flat [N*k]
  int* out_ctr = out_nbr + (size_t)n * K_NN;         // then center indices
  const int blocks = n / 16;                         // 16 queries per block
  knn_wmma_kernel<<<blocks, 256, 0, stream>>>(pos, n, out_nbr, out_ctr);
}